// PARTViT_61306363183642
// MI455X (gfx1250) — compile-verified
//
#include <hip/hip_runtime.h>

#define B_  64
#define N_  576
#define D_  768
#define K_  1024
#define NT  2

typedef __attribute__((ext_vector_type(2))) float v2f;
typedef __attribute__((ext_vector_type(8))) float v8f;

// ---------------------------------------------------------------------------
// Stage 1: G[m][j] = dot(z_flat[m, :], Wc[:, j])  for m in [0, B*N), j in 0..3
//   Wc[:,0:2] = W[0:768, :]   (applied to pair element 0)
//   Wc[:,2:4] = W[768:1536,:] (applied to pair element 1)
// One wave = one 16-row tile of z; V_WMMA_F32_16X16X4_F32 over K=768 (192
// steps, dual accumulators for matrix-pipe ILP).
// ---------------------------------------------------------------------------
__global__ void __launch_bounds__(256)
partvit_gemv_wmma(const float* __restrict__ z,
                  const float* __restrict__ W,
                  float* __restrict__ G)
{
    // LDS image: Wt[j][k] = Wc[k][j] for j<4, zero rows 4..15 (B-fragment
    // columns 4..15 must read as 0.0f). Branch-free, statically strided fill.
    __shared__ float Wt[16 * D_];
#pragma unroll
    for (int j = 0; j < 2; ++j)
        for (int k = threadIdx.x; k < D_; k += 256)
            Wt[j * D_ + k] = W[k * NT + j];              // top half of W
#pragma unroll
    for (int j = 2; j < 4; ++j)
        for (int k = threadIdx.x; k < D_; k += 256)
            Wt[j * D_ + k] = W[(D_ + k) * NT + (j - 2)]; // bottom half of W
    for (int i = threadIdx.x; i < 12 * D_; i += 256)
        Wt[4 * D_ + i] = 0.0f;                           // zero padding rows
    __syncthreads();

    const int warp = threadIdx.x >> 5;
    const int lane = threadIdx.x & 31;
    const int half = lane >> 4;     // selects K pair {k0,k0+1} vs {k0+2,k0+3}
    const int col  = lane & 15;     // A: row M ; B: column N
    const int tile = blockIdx.x * 8 + warp;      // 16-row tile index
    const int row  = tile * 16 + col;            // this lane's z row

    const float* zrow = z  + (size_t)row * D_ + 2 * half;
    const float* wrow = Wt + col * D_ + 2 * half;

    v8f c0 = {};
    v8f c1 = {};
    for (int k0 = 0; k0 < D_; k0 += 32) {
        // stream hint: ~512B ahead on this lane's z row (speculative, safe OOB)
        __builtin_prefetch(zrow + k0 + 128, 0, 3);
#pragma unroll
        for (int kk = 0; kk < 32; kk += 8) {
            v2f a0 = *(const v2f*)(zrow + k0 + kk);
            v2f b0 = *(const v2f*)(wrow + k0 + kk);
            c0 = __builtin_amdgcn_wmma_f32_16x16x4_f32(false, a0, false, b0,
                                                       (short)0, c0, false, false);
            v2f a1 = *(const v2f*)(zrow + k0 + kk + 4);
            v2f b1 = *(const v2f*)(wrow + k0 + kk + 4);
            c1 = __builtin_amdgcn_wmma_f32_16x16x4_f32(false, a1, false, b1,
                                                       (short)0, c1, false, false);
        }
    }
    v8f c = c0 + c1;   // disjoint-K partial sums

    // C/D layout: lane = N + 16*(M>=8); VGPR r holds M = r + 8*half; N = col.
    // Only columns 0..3 of C are meaningful -> store them to G[m*4 + j].
    if (col < 4) {
        float* g = G + (size_t)(tile * 16 + 8 * half) * 4 + col;
#pragma unroll
        for (int r = 0; r < 8; ++r)
            g[(size_t)r * 4] = c[r];
    }
}

// ---------------------------------------------------------------------------
// Stage 2: out[b,k,t] = G[b*N + i0][t] + G[b*N + i1][2+t] + bias[t]
// One thread per pair; ~1.5 MB of traffic total.
// ---------------------------------------------------------------------------
__global__ void __launch_bounds__(256)
partvit_gather(const int* __restrict__ idx,
               const float* __restrict__ G,
               const float* __restrict__ bias,
               float* __restrict__ out)
{
    int p  = blockIdx.x * 256 + threadIdx.x;   // pair id in [0, B*K)
    int b  = p >> 10;                          // K = 1024
    int i0 = idx[2 * p + 0];
    int i1 = idx[2 * p + 1];

    const float* g0 = G + (size_t)(b * N_ + i0) * 4;
    const float* g1 = G + (size_t)(b * N_ + i1) * 4;

    v2f r;
    r.x = g0[0] + g1[2] + bias[0];
    r.y = g0[1] + g1[3] + bias[1];
    *(v2f*)(out + 2 * p) = r;
}

extern "C" void kernel_launch(void* const* d_in, const int* in_sizes, int n_in,
                              void* d_out, int out_size, void* d_ws, size_t ws_size,
                              hipStream_t stream)
{
    const float* z    = (const float*)d_in[0];
    const int*   idx  = (const int*)d_in[1];
    const float* W    = (const float*)d_in[2];
    const float* bias = (const float*)d_in[3];

    float* G   = (float*)d_ws;        // B*N*4 floats = 576 KB scratch
    float* out = (float*)d_out;

    // Stage 1: 36864 rows / (16 rows/wave * 8 waves/block) = 288 blocks
    partvit_gemv_wmma<<<(B_ * N_) / (16 * 8), 256, 0, stream>>>(z, W, G);

    // Stage 2: 65536 pairs / 256 = 256 blocks
    partvit_gather<<<(B_ * K_) / 256, 256, 0, stream>>>(idx, G, bias, out);

    // Reference returns (out, indices): if d_out is the concatenated tuple,
    // echo indices after the float outputs.
    int out_elems = B_ * K_ * NT;
    if (out_size >= 2 * out_elems && n_in > 1) {
        hipMemcpyAsync((char*)d_out + (size_t)out_elems * sizeof(float),
                       d_in[1], (size_t)in_sizes[1] * sizeof(int),
                       hipMemcpyDeviceToDevice, stream);
    }
}